// StupidIndexator_26053271617888
// MI455X (gfx1250) — compile-verified
//
#include <hip/hip_runtime.h>
#include <math.h>

// V_WMMA_F32_16X16X4_F32 operand types (wave32):
//  A: 16x4 f32 = 64 vals / 32 lanes = 2 VGPRs -> float2 ext vector
//  B: 4x16 f32 = 64 vals             = 2 VGPRs -> float2 ext vector
//  C/D: 16x16 f32 = 256 vals         = 8 VGPRs -> float8 ext vector
typedef __attribute__((ext_vector_type(2))) float v2f;
typedef __attribute__((ext_vector_type(8))) float v8f;

#define BLOCK_THREADS 512          // 16 wave32 waves per block (one WGP-friendly size)
#define MAX_TILES     64           // stage up to 64*16 = 1024 hkl per LDS chunk (16 KB)
#define PHI_MAX_F     0.005f

// One block per candidate rotation n.
//  Phase A: thread 0 builds recip_rot = R(omega_n) @ reciprocal.
//  Phase B: all threads stage normalized theoretical rays into LDS in the
//           B-operand layout of V_WMMA_F32_16X16X4_F32:
//             lanes 0..15 : (g_x, g_y)/|g| for column N = lane   (K=0,1)
//             lanes 16..31: (g_z, 0 )/|g| for column N = lane-16 (K=2,3)
//  Phase C: wave w (= q-tile w) holds its A fragment in registers and sweeps
//           all m-tiles: one v_wmma_f32_16x16x4_f32 per tile, then elementwise
//           v_max into a running 8xVGPR max (this folds max-over-m).
//  Phase D: cross-lane max butterfly (16-lane halves), threshold count,
//           single store of rate[n].
__global__ __launch_bounds__(BLOCK_THREADS)
void laue_match_kernel(const float* __restrict__ u_q,
                       const float* __restrict__ angle,
                       const float* __restrict__ recip,   // 3x3 row-major
                       const int*   __restrict__ hkl,     // (m,3)
                       int n_spots, int n_hkl,
                       int* __restrict__ rate_out)
{
    __shared__ float R[9];                 // recip_rot, row-major
    __shared__ v2f   Bstage[MAX_TILES][32];
    __shared__ int   s_rate;

    const int n    = blockIdx.x;
    const int tid  = threadIdx.x;
    const int lane = tid & 31;
    const int wave = tid >> 5;

    if (tid == 0) {
        const float w1 = angle[3 * n + 0];
        const float w2 = angle[3 * n + 1];
        const float w3 = angle[3 * n + 2];
        const float c1 = cosf(w1), c2 = cosf(w2), c3 = cosf(w3);
        const float s1 = sinf(w1), s2 = sinf(w2), s3 = sinf(w3);
        float rot[9];
        rot[0] = c2 * c3;                 rot[1] = -c2 * s3;                rot[2] = s2;
        rot[3] = c1 * s3 + s1 * s2 * c3;  rot[4] = c1 * c3 - s1 * s2 * s3;  rot[5] = -s1 * c2;
        rot[6] = s1 * s3 - c1 * s2 * c3;  rot[7] = s1 * c3 + c1 * s2 * s3;  rot[8] = c1 * c2;
        // recip_rot[i][k] = sum_j rot[i][j] * recip[j][k]
        for (int i = 0; i < 3; ++i)
            for (int k = 0; k < 3; ++k)
                R[3 * i + k] = rot[3 * i + 0] * recip[0 * 3 + k]
                             + rot[3 * i + 1] * recip[1 * 3 + k]
                             + rot[3 * i + 2] * recip[2 * 3 + k];
        s_rate = 0;
    }
    __syncthreads();

    const float cosPhi = cosf(PHI_MAX_F);
    const int   qtiles = (n_spots + 15) >> 4;
    const bool  active = (wave < qtiles);   // wave-uniform: EXEC stays all-1s for WMMA

    // A fragment: rows = q spots of this wave's tile.
    v2f afrag; afrag.x = 0.0f; afrag.y = 0.0f;
    if (active) {
        const int row = wave * 16 + (lane & 15);
        if (row < n_spots) {
            if (lane < 16) { afrag.x = u_q[3 * row + 0]; afrag.y = u_q[3 * row + 1]; }
            else           { afrag.x = u_q[3 * row + 2]; afrag.y = 0.0f; }
        }
    }

    v8f vmax;
    #pragma unroll
    for (int i = 0; i < 8; ++i) vmax[i] = -2.0f;   // dots are cosines in [-1,1]

    for (int m0 = 0; m0 < n_hkl; m0 += MAX_TILES * 16) {
        const int chunk_m     = min(n_hkl - m0, MAX_TILES * 16);
        const int chunk_tiles = (chunk_m + 15) >> 4;

        __syncthreads();   // previous chunk's reads done before restaging
        for (int e = tid; e < chunk_tiles * 32; e += BLOCK_THREADS) {
            const int t  = e >> 5;
            const int l  = e & 31;
            const int mi = m0 + t * 16 + (l & 15);
            v2f b; b.x = 0.0f; b.y = 0.0f;     // padded columns -> dot 0 < cosPhi
            if (mi < n_hkl) {
                const float h0 = (float)hkl[3 * mi + 0];
                const float h1 = (float)hkl[3 * mi + 1];
                const float h2 = (float)hkl[3 * mi + 2];
                const float gx = R[0] * h0 + R[1] * h1 + R[2] * h2;
                const float gy = R[3] * h0 + R[4] * h1 + R[5] * h2;
                const float gz = R[6] * h0 + R[7] * h1 + R[8] * h2;
                const float inv = 1.0f / sqrtf(gx * gx + gy * gy + gz * gz);
                if (l < 16) { b.x = gx * inv; b.y = gy * inv; }
                else        { b.x = gz * inv; b.y = 0.0f;     }
            }
            Bstage[t][l] = b;
        }
        __syncthreads();

        if (active) {
            for (int t = 0; t < chunk_tiles; ++t) {
                const v2f bfrag = Bstage[t][lane];
                v8f c;
                #pragma unroll
                for (int i = 0; i < 8; ++i) c[i] = 0.0f;
                // D = A(16x4) x B(4x16) + 0 : 16x16 f32 tile of cos(phi)
                v8f d = __builtin_amdgcn_wmma_f32_16x16x4_f32(
                    /*neg_a=*/false, afrag, /*neg_b=*/false, bfrag,
                    /*c_mod=*/(short)0, c, /*reuse_a=*/false, /*reuse_b=*/false);
                #pragma unroll
                for (int i = 0; i < 8; ++i) vmax[i] = fmaxf(vmax[i], d[i]);
            }
        }
    }

    if (active) {
        // Row-max across the 16 columns (columns live across lanes; rows split
        // at lane 16: VGPR r -> M=r in lanes 0..15, M=8+r in lanes 16..31).
        float rm[8];
        #pragma unroll
        for (int i = 0; i < 8; ++i) {
            float v = vmax[i];
            v = fmaxf(v, __shfl_xor(v, 1, 32));
            v = fmaxf(v, __shfl_xor(v, 2, 32));
            v = fmaxf(v, __shfl_xor(v, 4, 32));
            v = fmaxf(v, __shfl_xor(v, 8, 32));
            rm[i] = v;
        }
        if (lane == 0 || lane == 16) {
            const int base = wave * 16 + ((lane == 16) ? 8 : 0);
            int cnt = 0;
            #pragma unroll
            for (int r = 0; r < 8; ++r) {
                const int q = base + r;
                if (q < n_spots && rm[r] >= cosPhi) ++cnt;
            }
            atomicAdd(&s_rate, cnt);
        }
    }
    __syncthreads();
    if (tid == 0) rate_out[n] = s_rate;
}

// Stable descending rank sort of 1024 (rate, idx) pairs, matching
// jnp.argsort(-rate) (ties broken by ascending index). O(n^2) is ~1M ops.
__global__ void rank_sort_kernel(const float* __restrict__ angle,
                                 const int*   __restrict__ rate,
                                 float* __restrict__ out_angle,
                                 int*   __restrict__ out_rate,
                                 int n_angles)
{
    const int i = blockIdx.x * blockDim.x + threadIdx.x;
    if (i >= n_angles) return;
    const int ri = rate[i];
    int rank = 0;
    for (int j = 0; j < n_angles; ++j) {
        const int rj = rate[j];
        if (rj > ri || (rj == ri && j < i)) ++rank;
    }
    out_angle[3 * rank + 0] = angle[3 * i + 0];
    out_angle[3 * rank + 1] = angle[3 * i + 1];
    out_angle[3 * rank + 2] = angle[3 * i + 2];
    out_rate[rank] = ri;
}

extern "C" void kernel_launch(void* const* d_in, const int* in_sizes, int n_in,
                              void* d_out, int out_size, void* d_ws, size_t ws_size,
                              hipStream_t stream)
{
    const float* u_q   = (const float*)d_in[0];   // (q,3) f32
    const float* angle = (const float*)d_in[1];   // (n,3) f32
    const float* recip = (const float*)d_in[2];   // (3,3) f32
    const int*   hkl   = (const int*)d_in[3];     // (m,3) i32

    const int n_spots  = in_sizes[0] / 3;
    const int n_angles = in_sizes[1] / 3;
    const int n_hkl    = in_sizes[3] / 3;

    int* rate_ws = (int*)d_ws;                    // n_angles ints of scratch

    laue_match_kernel<<<n_angles, BLOCK_THREADS, 0, stream>>>(
        u_q, angle, recip, hkl, n_spots, n_hkl, rate_ws);

    float* out_angle = (float*)d_out;             // first 3*n floats
    int*   out_rate  = (int*)d_out + 3 * n_angles; // then n int32 counts

    rank_sort_kernel<<<(n_angles + 255) / 256, 256, 0, stream>>>(
        angle, rate_ws, out_angle, out_rate, n_angles);
}